// RGCN_70660801954147
// MI455X (gfx1250) — compile-verified
//
#include <hip/hip_runtime.h>
#include <hip/hip_bf16.h>

#define N_NODES 50000
#define N_EDGES 300000
#define N_REL   8
#define DIM     256
#define MB      5                       // M-tiles per wave (50000 = 625 * 80)
#define GEMM_BLOCKS (N_NODES / (16 * MB))  // 625, exact

typedef __attribute__((ext_vector_type(2))) float v2f;
typedef __attribute__((ext_vector_type(8))) float v8f;

// ---------------------------------------------------------------------------
// GEMM: C (+)= A[50000x256] * B[256x256]  (optionally C initialized from bias)
// Block = 128 threads = 4 waves. Block covers rows [blockIdx.x*80, +80);
// wave w covers cols [w*64, +64). Each wave holds 5 (M) x 4 (N) 16x16 f32
// accumulators; per K-step it loads B fragments ONCE and reuses them across
// all 5 M-tiles: 13 loads -> 20 v_wmma_f32_16x16x4_f32.
//
// Fragment layouts per CDNA5 ISA 7.12.2 (wave32):
//   A 16x4 f32 : lane<16 -> row=lane,    holds K=k0+0,k0+1
//                lane>=16 -> row=lane-16, holds K=k0+2,k0+3
//   B 4x16 f32 : VGPR0 = row k0+{0|2} (lane half), VGPR1 = next row, col=lane&15
//   C 16x16 f32: component j -> row = j + (lane>=16 ? 8 : 0), col = lane&15
// ---------------------------------------------------------------------------
__global__ __launch_bounds__(128)
void rgcn_gemm(const float* __restrict__ A, const float* __restrict__ B,
               const float* __restrict__ bias, float* __restrict__ C,
               int init_from_bias)
{
    const int wave    = threadIdx.x >> 5;
    const int lane    = threadIdx.x & 31;
    const int hi      = lane >> 4;        // 0: lanes 0-15, 1: lanes 16-31
    const int ln      = lane & 15;
    const int mBase   = blockIdx.x * (16 * MB);
    const int colBase = wave * 64;

    v8f acc[MB][4];
    if (init_from_bias) {
        #pragma unroll
        for (int t = 0; t < 4; ++t) {
            const float bv = bias[colBase + t * 16 + ln];
            #pragma unroll
            for (int m = 0; m < MB; ++m)
                #pragma unroll
                for (int j = 0; j < 8; ++j) acc[m][t][j] = bv;
        }
    } else {
        #pragma unroll
        for (int m = 0; m < MB; ++m)
            #pragma unroll
            for (int t = 0; t < 4; ++t)
                #pragma unroll
                for (int j = 0; j < 8; ++j)
                    acc[m][t][j] = C[(size_t)(mBase + m * 16 + hi * 8 + j) * DIM
                                     + colBase + t * 16 + ln];
    }

    const float* aBase = A + (size_t)(mBase + ln) * DIM + hi * 2;

    #pragma unroll 2
    for (int k0 = 0; k0 < DIM; k0 += 4) {
        // B fragments: loaded once, reused by all MB M-tiles
        const float* bPtr = B + (size_t)(k0 + hi * 2) * DIM + colBase + ln;
        v2f b[4];
        #pragma unroll
        for (int t = 0; t < 4; ++t) {
            b[t].x = bPtr[t * 16];         // row k0 + hi*2
            b[t].y = bPtr[t * 16 + DIM];   // row k0 + hi*2 + 1
        }
        // A fragments: one b64 per M-tile (immediate row offsets)
        v2f a[MB];
        #pragma unroll
        for (int m = 0; m < MB; ++m)
            a[m] = *(const v2f*)(aBase + (size_t)m * 16 * DIM + k0);

        #pragma unroll
        for (int m = 0; m < MB; ++m)
            #pragma unroll
            for (int t = 0; t < 4; ++t)
                acc[m][t] = __builtin_amdgcn_wmma_f32_16x16x4_f32(
                    false, a[m], false, b[t], (short)0, acc[m][t], false, false);
    }

    #pragma unroll
    for (int m = 0; m < MB; ++m)
        #pragma unroll
        for (int t = 0; t < 4; ++t)
            #pragma unroll
            for (int j = 0; j < 8; ++j)
                C[(size_t)(mBase + m * 16 + hi * 8 + j) * DIM
                  + colBase + t * 16 + ln] = acc[m][t][j];
}

// ---------------------------------------------------------------------------
// Per-relation scatter: agg[dst] += h[src] for edges with edge_type == rel.
// 64 threads per edge, float4 gather + 4x global_atomic_add_f32.
// ---------------------------------------------------------------------------
__global__ __launch_bounds__(256)
void scatter_rel(const float* __restrict__ h, const int* __restrict__ ei,
                 const int* __restrict__ et, float* __restrict__ agg, int rel)
{
    const long gid = (long)blockIdx.x * blockDim.x + threadIdx.x;
    const long e   = gid >> 6;
    if (e >= N_EDGES) return;
    if (et[e] != rel) return;
    const int part = (int)(gid & 63) << 2;     // 0..252 step 4
    const int s = ei[e];
    const int d = ei[N_EDGES + e];
    const float4 v = *(const float4*)(h + (size_t)s * DIM + part);
    float* p = agg + (size_t)d * DIM + part;
    atomicAdd(p + 0, v.x);
    atomicAdd(p + 1, v.y);
    atomicAdd(p + 2, v.z);
    atomicAdd(p + 3, v.w);
}

__global__ __launch_bounds__(256)
void fill_zero4(float4* __restrict__ p, int n4)
{
    const int i = blockIdx.x * blockDim.x + threadIdx.x;
    if (i < n4) p[i] = make_float4(0.f, 0.f, 0.f, 0.f);
}

// mode 0 = relu, 1 = sigmoid (in place)
__global__ __launch_bounds__(256)
void activation4(float4* __restrict__ p, int n4, int mode)
{
    const int i = blockIdx.x * blockDim.x + threadIdx.x;
    if (i >= n4) return;
    float4 v = p[i];
    if (mode == 0) {
        v.x = fmaxf(v.x, 0.f); v.y = fmaxf(v.y, 0.f);
        v.z = fmaxf(v.z, 0.f); v.w = fmaxf(v.w, 0.f);
    } else {
        v.x = 1.f / (1.f + __expf(-v.x));
        v.y = 1.f / (1.f + __expf(-v.y));
        v.z = 1.f / (1.f + __expf(-v.z));
        v.w = 1.f / (1.f + __expf(-v.w));
    }
    p[i] = v;
}

extern "C" void kernel_launch(void* const* d_in, const int* in_sizes, int n_in,
                              void* d_out, int out_size, void* d_ws, size_t ws_size,
                              hipStream_t stream)
{
    const float* x      = (const float*)d_in[0];   // [50000,256]
    const float* W      = (const float*)d_in[1];   // [3,8,256,256]
    const float* roots  = (const float*)d_in[2];   // [3,256,256]
    const float* biases = (const float*)d_in[3];   // [3,256]
    const int*   ei     = (const int*)d_in[4];     // [2,300000]
    const int*   et     = (const int*)d_in[5];     // [300000]
    float* out = (float*)d_out;

    const size_t HSZ = (size_t)N_NODES * DIM;      // 12.8M floats
    float* hA  = (float*)d_ws;                     // layer-0 output
    float* hB  = hA + HSZ;                         // layer-1 output
    float* agg = hB + HSZ;                         // per-relation aggregate

    const int n4 = (int)(HSZ / 4);                 // 3.2M float4
    const int fillBlocks    = (n4 + 255) / 256;
    const int scatterBlocks = (int)(((long)N_EDGES * 64 + 255) / 256);

    const float* src_h = x;
    for (int l = 0; l < 3; ++l) {
        float* dst = (l == 0) ? hA : (l == 1) ? hB : out;

        // dst = src_h @ roots[l] + bias[l]
        rgcn_gemm<<<GEMM_BLOCKS, 128, 0, stream>>>(
            src_h, roots + (size_t)l * DIM * DIM, biases + (size_t)l * DIM, dst, 1);

        for (int r = 0; r < N_REL; ++r) {
            fill_zero4<<<fillBlocks, 256, 0, stream>>>((float4*)agg, n4);
            scatter_rel<<<scatterBlocks, 256, 0, stream>>>(src_h, ei, et, agg, r);
            // dst += agg @ W[l][r]
            rgcn_gemm<<<GEMM_BLOCKS, 128, 0, stream>>>(
                agg, W + (size_t)(l * N_REL + r) * DIM * DIM, nullptr, dst, 0);
        }

        activation4<<<fillBlocks, 256, 0, stream>>>((float4*)dst, n4, (l == 2) ? 1 : 0);
        src_h = dst;
    }
}